// FisherVectorLayer_6837587935988
// MI455X (gfx1250) — compile-verified
//
#include <hip/hip_runtime.h>
#include <math.h>

// ---------------- problem constants ----------------
#define B_ 16
#define N_ 16384
#define D_ 64
#define K_ 64
#define TILE_N 128
#define NTILES (N_ / TILE_N)   // 128 tiles per batch
// LDS padding strides (floats) to avoid bank conflicts
#define SXS 68   // x tile row stride: 272B rows -> 16B aligned, bank-staggered
#define SQS 65   // Q / logit tile row stride
#define SWS 65   // weight tile row stride

typedef __attribute__((ext_vector_type(2))) float v2f;
typedef __attribute__((ext_vector_type(8))) float v8f;

// =====================================================================
// Prep: build fused weight matrices in [d][k] layout (natural B-fragment
// order), fold all per-k constants into bias[k], zero the accumulators.
//   logit[n,k] = bias[k] + dot(x^2, Wa[:,k]) + dot(x, Wb[:,k])
//   Wa[d][k] = -0.5 / var[k][d]
//   Wb[d][k] = mu[k][d] / var[k][d]
//   bias[k]  = log(pi[k]) - 0.5*(D*log(2pi) + logdet[k] + mu^T inv_var mu)
// =====================================================================
__global__ __launch_bounds__(256) void fv_prep(
    const float* __restrict__ pi, const float* __restrict__ mu,
    const float* __restrict__ var,
    float* __restrict__ Wa, float* __restrict__ Wb, float* __restrict__ bias,
    float* __restrict__ accS, float* __restrict__ accX, float* __restrict__ accX2)
{
  const int gid = blockIdx.x * blockDim.x + threadIdx.x;   // 65536 threads
  accX[gid]  = 0.0f;                                       // B*K*D == 65536
  accX2[gid] = 0.0f;
  if (gid < B_ * K_) accS[gid] = 0.0f;
  if (gid < D_ * K_) {
    const int d = gid >> 6, k = gid & 63;
    const float iv = 1.0f / var[k * D_ + d];
    Wa[gid] = -0.5f * iv;
    Wb[gid] = mu[k * D_ + d] * iv;
  }
  if (gid < K_) {
    float logdet = 0.0f, m2 = 0.0f;
    for (int d = 0; d < D_; ++d) {
      const float v = var[gid * D_ + d];
      const float m = mu[gid * D_ + d];
      logdet += __logf(v);
      m2     += m * m / v;
    }
    // log(2*pi) = 1.8378770664093453
    bias[gid] = __logf(pi[gid]) - 0.5f * ((float)D_ * 1.8378770664093453f + logdet + m2);
  }
}

// =====================================================================
// Pass 1 (streaming, one HBM read of x):
//  1. async-copy 128x64 x-tile global->LDS (ASYNCcnt path, no VGPR staging)
//  2. WMMA f32 16x16x4 logit GEMM (x^2*Wa + x*Wb + bias)
//  3. softmax over K in LDS -> Q
//  4. column sums of Q -> atomic accS
//  5. WMMA f32 Q^T * [x, x^2] -> atomic accX / accX2  (L2-resident, 512KB)
// =====================================================================
__global__ __launch_bounds__(256) void fv_pass1(
    const float* __restrict__ x,
    const float* __restrict__ Wa, const float* __restrict__ Wb,
    const float* __restrict__ bias,
    float* __restrict__ accS, float* __restrict__ accX, float* __restrict__ accX2)
{
  __shared__ float sx[TILE_N][SXS];   // x tile        (~34.8 KB)
  __shared__ float sq[TILE_N][SQS];   // logits -> Q   (~33.3 KB)
  __shared__ float swa[D_][SWS];      // Wa [d][k]     (~16.6 KB)
  __shared__ float swb[D_][SWS];      // Wb [d][k]     (~16.6 KB)
  __shared__ float sbias[K_];

  const int tid  = threadIdx.x;
  const int lane = tid & 31;          // wave32
  const int wave = tid >> 5;          // 8 waves / block
  const int b    = blockIdx.x / NTILES;
  const int n0   = (blockIdx.x % NTILES) * TILE_N;
  const float* xb = x + ((size_t)b * N_ + n0) * D_;

  // ---- stage x tile via async global->LDS (16B per lane per issue) ----
  // GVS mode: SGPR64 base + per-lane 32-bit offset. Both global rows (256B)
  // and padded LDS rows (272B) keep 16B alignment for B128 transfers.
  {
    const unsigned long long gbase = (unsigned long long)xb;
    for (int i = tid; i < TILE_N * (D_ / 4); i += 256) {
      const int row = i >> 4;              // 16 float4 per row
      const int c4  = (i & 15) * 4;
      const unsigned ldsaddr = (unsigned)(unsigned long long)(void*)&sx[row][c4];
      const unsigned gofs    = (unsigned)((row * D_ + c4) * 4);
      asm volatile("global_load_async_to_lds_b128 %0, %1, %2"
                   :: "v"(ldsaddr), "v"(gofs), "s"(gbase)
                   : "memory");
    }
    asm volatile("s_wait_asynccnt 0" ::: "memory");
  }

  // ---- stage weights + bias (tiny, L2-resident) ----
  for (int i = tid; i < D_ * K_; i += 256) {
    const int d = i >> 6, k = i & 63;
    swa[d][k] = Wa[i];
    swb[d][k] = Wb[i];
  }
  if (tid < K_) sbias[tid] = bias[tid];
  __syncthreads();

  const int r16 = lane & 15;
  const int hi  = lane >> 4;          // 0: K-dim 0/1, 1: K-dim 2/3 (A frag halves)

  // ---------------- GEMM1: logits ----------------
  // 32 tiles of 16(n) x 16(k); 4 per wave; K-dim = D (x2 chain + x chain)
  for (int t = wave; t < (TILE_N / 16) * (K_ / 16); t += 8) {
    const int nt = t >> 2, kt = t & 3;
    const float bv = sbias[kt * 16 + r16];      // bias depends on k (= N col) only
    v8f c = {bv, bv, bv, bv, bv, bv, bv, bv};
    const int arow = nt * 16 + r16;             // A: M rows = n
    const int kcol = kt * 16 + r16;             // B: N cols = k
#pragma unroll
    for (int d = 0; d < D_; d += 4) {
      const int dd = d + 2 * hi;
      v2f ax;  ax.x  = sx[arow][dd];    ax.y  = sx[arow][dd + 1];
      v2f a2;  a2.x  = ax.x * ax.x;     a2.y  = ax.y * ax.y;
      v2f bwa; bwa.x = swa[dd][kcol];   bwa.y = swa[dd + 1][kcol];
      v2f bwb; bwb.x = swb[dd][kcol];   bwb.y = swb[dd + 1][kcol];
      c = __builtin_amdgcn_wmma_f32_16x16x4_f32(false, a2, false, bwa, (short)0, c, false, false);
      c = __builtin_amdgcn_wmma_f32_16x16x4_f32(false, ax, false, bwb, (short)0, c, false, false);
    }
#pragma unroll
    for (int r = 0; r < 8; ++r)                 // C: M = r + 8*hi, N = r16
      sq[nt * 16 + r + 8 * hi][kt * 16 + r16] = c[r];
  }
  __syncthreads();

  // ---------------- softmax over K per row ----------------
  if (tid < TILE_N) {
    float mx = -3.4e38f;
#pragma unroll 8
    for (int k = 0; k < K_; ++k) mx = fmaxf(mx, sq[tid][k]);
    float s = 0.0f;
#pragma unroll 8
    for (int k = 0; k < K_; ++k) {
      const float e = __expf(sq[tid][k] - mx);
      s += e;
      sq[tid][k] = e;
    }
    const float inv = 1.0f / s;
#pragma unroll 8
    for (int k = 0; k < K_; ++k) sq[tid][k] *= inv;
  }
  __syncthreads();

  // ---------------- Q column sums -> accS ----------------
  {
    const int k = tid & 63;
    const int part = tid >> 6;                  // 4 partials of 32 rows
    float s = 0.0f;
#pragma unroll 8
    for (int r = part * 32; r < part * 32 + 32; ++r) s += sq[r][k];
    __hip_atomic_fetch_add(&accS[b * K_ + k], s, __ATOMIC_RELAXED,
                           __HIP_MEMORY_SCOPE_AGENT);
  }

  // ---------------- GEMM2: Q^T * [x, x2] ----------------
  // 32 tiles: which(2) x kt(4) x dt(4); 4 per wave; K-dim = TILE_N
  for (int t = wave; t < 32; t += 8) {
    const int which = t >> 4;                   // 0: x, 1: x^2
    const int kt = (t >> 2) & 3;
    const int dt = t & 3;
    v8f c = {0, 0, 0, 0, 0, 0, 0, 0};
    const int mcol = kt * 16 + r16;             // A: M rows = k (Q transposed)
    const int dcol = dt * 16 + r16;             // B: N cols = d
#pragma unroll 8
    for (int nn = 0; nn < TILE_N; nn += 4) {
      const int rb = nn + 2 * hi;
      v2f a;  a.x  = sq[rb][mcol];  a.y  = sq[rb + 1][mcol];
      v2f bb; bb.x = sx[rb][dcol];  bb.y = sx[rb + 1][dcol];
      if (which) { bb.x *= bb.x; bb.y *= bb.y; }   // wave-uniform branch
      c = __builtin_amdgcn_wmma_f32_16x16x4_f32(false, a, false, bb, (short)0, c, false, false);
    }
    float* acc = which ? accX2 : accX;
#pragma unroll
    for (int r = 0; r < 8; ++r) {
      const int kg = kt * 16 + r + 8 * hi;
      __hip_atomic_fetch_add(&acc[((size_t)b * K_ + kg) * D_ + dcol], c[r],
                             __ATOMIC_RELAXED, __HIP_MEMORY_SCOPE_AGENT);
    }
  }
}

// =====================================================================
// Finalize: Fisher-vector statistics from the accumulators.
// out[b] = [ d_pi (K) | d_mu (K*D) | d_sigma (K*D) ]
// =====================================================================
__global__ __launch_bounds__(64) void fv_finalize(
    const float* __restrict__ pi, const float* __restrict__ mu,
    const float* __restrict__ var,
    const float* __restrict__ accS, const float* __restrict__ accX,
    const float* __restrict__ accX2, float* __restrict__ out)
{
  const int bk = blockIdx.x;               // B*K blocks
  const int b = bk >> 6, k = bk & 63;
  const int d = threadIdx.x;
  const float invN = 1.0f / (float)N_;
  const float qs  = accS[bk] * invN;
  const float qx  = accX [(size_t)bk * D_ + d] * invN;
  const float qx2 = accX2[(size_t)bk * D_ + d] * invN;
  const float m = mu[k * D_ + d];
  const float v = var[k * D_ + d];
  float* ob = out + (size_t)b * (K_ + 2 * K_ * D_);
  ob[K_ + k * D_ + d]            = qx - qs * m;
  ob[K_ + K_ * D_ + k * D_ + d]  = -qx2 - qs * m * m + qs * v + 2.0f * qx * m;
  if (d == 0) ob[k] = qs - pi[k];
}

// =====================================================================
extern "C" void kernel_launch(void* const* d_in, const int* in_sizes, int n_in,
                              void* d_out, int out_size, void* d_ws, size_t ws_size,
                              hipStream_t stream) {
  const float* x   = (const float*)d_in[0];
  const float* pi  = (const float*)d_in[1];
  const float* mu  = (const float*)d_in[2];
  const float* var = (const float*)d_in[3];

  float* ws   = (float*)d_ws;
  float* Wa   = ws;                   //  4096 floats
  float* Wb   = Wa + D_ * K_;         //  4096
  float* bias = Wb + D_ * K_;         //    64
  float* accS = bias + K_;            //  1024
  float* accX = accS + B_ * K_;       // 65536
  float* accX2 = accX + B_ * K_ * D_; // 65536   (total ~561 KB of d_ws)

  fv_prep<<<256, 256, 0, stream>>>(pi, mu, var, Wa, Wb, bias, accS, accX, accX2);
  fv_pass1<<<B_ * NTILES, 256, 0, stream>>>(x, Wa, Wb, bias, accS, accX, accX2);
  fv_finalize<<<B_ * K_, 64, 0, stream>>>(pi, mu, var, accS, accX, accX2, (float*)d_out);
}